// GunnarODE_60017872994363
// MI455X (gfx1250) — compile-verified
//
#include <hip/hip_runtime.h>
#include <math.h>

// Problem constants (match reference)
#define L_T 128
#define B_T 256
#define H_T 256
#define C_T 17
#define Y_T 16
#define NHC (H_T * C_T)   /* 4352 */
#define GRPC (16 * C_T)   /* 272 columns per fused block = exactly 16 h-groups */
#define PADA 2            /* LDS pad so A-fragment reads hit distinct banks */

typedef float v2f __attribute__((ext_vector_type(2)));
typedef float v8f __attribute__((ext_vector_type(8)));

__device__ __forceinline__ v8f wmma_f32_k4(v2f a, v2f b, v8f c) {
    return __builtin_amdgcn_wmma_f32_16x16x4_f32(
        /*neg_a=*/false, a, /*neg_b=*/false, b,
        /*c_mod=*/(short)0, c, /*reuse_a=*/false, /*reuse_b=*/false);
}

// ---------------------------------------------------------------------------
// Hermite coefficient prep: one thread per (l, b, c), l in [0, L-1).
// Folds the step scale h/N_SUB = h/2 into the stored dX values.
// ---------------------------------------------------------------------------
__global__ void gunnar_prep(const float* __restrict__ ts,
                            const float* __restrict__ us,
                            float* __restrict__ dX0s,
                            float* __restrict__ dX1s)
{
    int tid = blockIdx.x * blockDim.x + threadIdx.x;
    const int total = (L_T - 1) * B_T * C_T;
    if (tid >= total) return;
    int l   = tid / (B_T * C_T);
    int rem = tid - l * (B_T * C_T);
    int b   = rem / C_T;
    int c   = rem - b * C_T;

    float t0 = ts[l * B_T];
    float t1 = ts[(l + 1) * B_T];
    float h  = t1 - t0;

    float xi, xn, xprev = 0.0f;
    if (c == 0) {
        xi = t0; xn = t1;
        if (l > 0) xprev = ts[(l - 1) * B_T];
    } else {
        xi = us[(l * B_T + b) * (C_T - 1) + (c - 1)];
        xn = us[((l + 1) * B_T + b) * (C_T - 1) + (c - 1)];
        if (l > 0) xprev = us[((l - 1) * B_T + b) * (C_T - 1) + (c - 1)];
    }

    float slope = (xn - xi) / h;
    float mi;
    if (l == 0) {
        mi = slope;
    } else {
        float hp = t0 - ts[(l - 1) * B_T];
        mi = (xi - xprev) / hp;
    }
    float inv_h  = 1.0f / h;
    float inv_h2 = inv_h * inv_h;
    float c2 = 3.0f * slope * inv_h - (2.0f * mi + slope) * inv_h;
    float c3 = (mi + slope) * inv_h2 - 2.0f * slope * inv_h2;
    float u  = 0.5f * h;
    float dX1 = mi + 2.0f * c2 * u + 3.0f * c3 * u * u;

    float s = 0.5f * h;          // h / N_SUB
    dX0s[tid] = mi  * s;
    dX1s[tid] = dX1 * s;
}

// z0 = zeros @ enc_W + enc_b  ==  enc_b broadcast over batch
__global__ void gunnar_init_z(const float* __restrict__ enc_b,
                              float* __restrict__ z)
{
    int tid = blockIdx.x * blockDim.x + threadIdx.x;
    if (tid < B_T * H_T) z[tid] = enc_b[tid & (H_T - 1)];
}

// ---------------------------------------------------------------------------
// Generic fp32 WMMA GEMM with compile-time dims:
//   D = act(A(MxK) * B(KxN) + bias), row-major.  One wave per 16x16 tile,
//   V_WMMA_F32_16X16X4_F32 over K/4 chunks.  ACT: 0 relu, 1 tanh, 2 none.
// Lane mapping per CDNA5 ISA 7.12.2 (32-bit operands):
//   n = lane%16, half = lane/16
//   A: lane holds A[n, kb..kb+1], kb = k0 + 2*half  (contiguous b64)
//   B: lane holds B[kb..kb+1, col0+n]
//   D: VGPR v -> row (row0 + v + 8*half), col (col0 + n)
// ---------------------------------------------------------------------------
template <int N, int K, int ACT>
__global__ void gunnar_gemm(const float* __restrict__ A,
                            const float* __restrict__ Bm,
                            const float* __restrict__ bias,
                            float* __restrict__ D)
{
    const int lane = threadIdx.x & 31;
    const int n    = lane & 15;
    const int half = lane >> 4;
    const int col0 = (blockIdx.x * blockDim.y + threadIdx.y) << 4;
    const int row0 = blockIdx.y << 4;
    if (col0 >= N) return;              // wave-uniform guard: EXEC all-1s at wmma

    v8f acc = {};
    const float* __restrict__ arow = A  + (row0 + n) * K + (half << 1);
    const float* __restrict__ bp   = Bm + (half << 1) * N + col0 + n;
    for (int k0 = 0; k0 < K; k0 += 4) {
        v2f a = *(const v2f*)(arow + k0);
        v2f b; b.x = bp[0]; b.y = bp[N];
        acc = wmma_f32_k4(a, b, acc);
        bp += 4 * N;
    }
    const float bz = bias[col0 + n];
    #pragma unroll
    for (int v = 0; v < 8; ++v) {
        float r = acc[v] + bz;
        if (ACT == 0)      r = fmaxf(r, 0.0f);
        else if (ACT == 1) r = tanhf(r);
        D[(row0 + v + (half << 3)) * N + col0 + n] = r;
    }
}

// ---------------------------------------------------------------------------
// Fused: g = tanh(hdn @ f_W2 + f_b2)  (tile in LDS, never hits global), then
//        z[b, h] += sum_c g[b, h*C + c] * dX[b, c]
// Block = (32, 17) threads: 17 waves, each one 16x16 tile of the 16(rows) x
// 272(cols) group; 272 = 16 h-groups * 17 channels, so the contraction is
// block-local.  Grid = (NHC/272 = 16, B/16 = 16).
// ---------------------------------------------------------------------------
__global__ __launch_bounds__(544)
void gunnar_fused_fz(const float* __restrict__ hdn,
                     const float* __restrict__ W2,
                     const float* __restrict__ b2,
                     const float* __restrict__ dXs,
                     float* __restrict__ z)
{
    __shared__ float ahdn[16][H_T + PADA];   // hdn slab (padded: stride 258)
    __shared__ float gtile[16][GRPC + 1];    // tanh(g) tile (stride 273, odd)
    __shared__ float dxr[16][C_T];           // dX slab

    const int tid  = threadIdx.y * 32 + threadIdx.x;
    const int grp  = blockIdx.x;             // h-range [grp*16, grp*16+16)
    const int row0 = blockIdx.y << 4;        // batch slab

    // Stage hdn[row0:row0+16, :] and dX[row0:row0+16, :] into LDS
    for (int i = tid; i < 16 * H_T; i += 544) {
        int r = i >> 8, c = i & (H_T - 1);
        ahdn[r][c] = hdn[(row0 + r) * H_T + c];
    }
    if (tid < 16 * C_T) {
        int r = tid / C_T, c = tid - r * C_T;
        dxr[r][c] = dXs[(row0 + r) * C_T + c];
    }
    __syncthreads();

    const int lane = threadIdx.x;
    const int n    = lane & 15;
    const int half = lane >> 4;
    const int colg = threadIdx.y << 4;             // local col tile base (0..256)
    const int col  = grp * GRPC + colg + n;        // global column in [0, 4352)

    v8f acc = {};
    const float* __restrict__ bp = W2 + (size_t)(half << 1) * NHC + col;
    for (int k0 = 0; k0 < H_T; k0 += 4) {
        v2f a = *(const v2f*)(&ahdn[n][k0 + (half << 1)]);   // ds_load_b64
        v2f b; b.x = bp[0]; b.y = bp[NHC];
        acc = wmma_f32_k4(a, b, acc);
        bp += 4 * (size_t)NHC;
    }
    const float bz = b2[col];
    #pragma unroll
    for (int v = 0; v < 8; ++v)
        gtile[v + (half << 3)][colg + n] = tanhf(acc[v] + bz);
    __syncthreads();

    // Contraction: 256 threads, one (row, h) each; 17 fmas from LDS
    if (tid < 256) {
        int r = tid >> 4, hloc = tid & 15;
        const float* __restrict__ gr = &gtile[r][hloc * C_T];
        const float* __restrict__ dx = dxr[r];
        float s = 0.0f;
        #pragma unroll
        for (int c = 0; c < C_T; ++c) s = fmaf(gr[c], dx[c], s);
        z[(row0 + r) * H_T + (grp << 4) + hloc] += s;
    }
}

// ---------------------------------------------------------------------------
extern "C" void kernel_launch(void* const* d_in, const int* in_sizes, int n_in,
                              void* d_out, int out_size, void* d_ws, size_t ws_size,
                              hipStream_t stream) {
    const float* ts    = (const float*)d_in[0];
    const float* us    = (const float*)d_in[1];
    /* d_in[2] = ys (unused), d_in[3] = batch_size (compile-time B_T)     */
    /* d_in[4] = enc_W: multiplied by exact zeros in reference -> unused  */
    const float* enc_b = (const float*)d_in[5];
    const float* dec_W = (const float*)d_in[6];
    const float* dec_b = (const float*)d_in[7];
    const float* f_W1  = (const float*)d_in[8];
    const float* f_b1  = (const float*)d_in[9];
    const float* f_W2  = (const float*)d_in[10];
    const float* f_b2  = (const float*)d_in[11];
    float* out = (float*)d_out;

    float* ws   = (float*)d_ws;
    float* z    = ws;                               // B*H
    float* hdn  = z    + (size_t)B_T * H_T;         // B*H
    float* dX0s = hdn  + (size_t)B_T * H_T;         // (L-1)*B*C
    float* dX1s = dX0s + (size_t)(L_T - 1) * B_T * C_T;

    const dim3 wblk(32, 4);
    const dim3 gridA(H_T / 64, B_T / 16);           // (4, 16)  z @ f_W1
    const dim3 gridD(1, B_T / 16);                  // (1, 16)  z @ dec_W (guarded)
    const dim3 fblk(32, 17);                        // 17 waves
    const dim3 gridF(NHC / GRPC, B_T / 16);         // (16, 16)

    // 1) Hermite coefficients for all steps/substeps
    {
        int total = (L_T - 1) * B_T * C_T;
        gunnar_prep<<<(total + 255) / 256, 256, 0, stream>>>(ts, us, dX0s, dX1s);
    }
    // 2) z0 = enc_b
    gunnar_init_z<<<(B_T * H_T) / 256, 256, 0, stream>>>(enc_b, z);
    // 3) out[0] = z0 @ dec_W + dec_b
    gunnar_gemm<Y_T, H_T, 2><<<gridD, wblk, 0, stream>>>(z, dec_W, dec_b, out);
    // 4) scan over 127 steps, 2 substeps each
    for (int l = 0; l < L_T - 1; ++l) {
        const float* dxs[2] = { dX0s + (size_t)l * B_T * C_T,
                                dX1s + (size_t)l * B_T * C_T };
        for (int j = 0; j < 2; ++j) {
            gunnar_gemm<H_T, H_T, 0><<<gridA, wblk, 0, stream>>>(z, f_W1, f_b1, hdn);
            gunnar_fused_fz<<<gridF, fblk, 0, stream>>>(hdn, f_W2, f_b2, dxs[j], z);
        }
        gunnar_gemm<Y_T, H_T, 2><<<gridD, wblk, 0, stream>>>(
            z, dec_W, dec_b, out + (size_t)(l + 1) * B_T * Y_T);
    }
}